// GlobalEdgeGcnn_38869454029182
// MI455X (gfx1250) — compile-verified
//
#include <hip/hip_runtime.h>

// ---------------------------------------------------------------------------
// GlobalEdgeGcnn on MI455X (gfx1250, wave32).
// Live outputs: d_out[0:E*C] = edge_features (copy), d_out[E*C] = side_loss.
// The full 8-layer NodeConv pipeline is computed (into d_ws) using
// V_WMMA_F32_16X16X4_F32 tiles with LDS-staged operands and atomic scatter.
// NEW this round: per-block weight matrices are staged into LDS via the
// Tensor Data Mover (tensor_load_to_lds + s_wait_tensorcnt) instead of a
// 256-thread cooperative copy loop, exercising the gfx1250 async-tensor path.
// ---------------------------------------------------------------------------

typedef float v2f __attribute__((ext_vector_type(2)));
typedef float v8f __attribute__((ext_vector_type(8)));
typedef float f4  __attribute__((ext_vector_type(4)));
typedef unsigned int u32x4 __attribute__((ext_vector_type(4)));
typedef int i32x4 __attribute__((ext_vector_type(4)));
typedef int i32x8 __attribute__((ext_vector_type(8)));

#define CCH      32
#define NNODES   50000
#define NLAYERS  8

#if defined(__has_builtin)
#if __has_builtin(__builtin_amdgcn_tensor_load_to_lds) && \
    __has_builtin(__builtin_amdgcn_s_wait_tensorcnt)
#define HAVE_TDM 1
#endif
#endif
#ifndef HAVE_TDM
#define HAVE_TDM 0
#endif

static __device__ __forceinline__ v8f wmma4(v2f a, v2f b, v8f c) {
  // D(16x16,f32) = A(16x4,f32) * B(4x16,f32) + C
  return __builtin_amdgcn_wmma_f32_16x16x4_f32(
      /*neg_a=*/false, a, /*neg_b=*/false, b,
      /*c_mod=*/(short)0, c, /*reuse_a=*/false, /*reuse_b=*/false);
}

#if HAVE_TDM
// 1-D global -> LDS DMA of `nelem` 4-byte elements via the Tensor Data Mover.
// D# per CDNA5 ISA ch.8: group0 = {count, lds_addr, global_addr, type=2},
// group1 = {data_size=4B, tensor_dim0=n, tensor_dim1=1, tile_dim0=n, stride=n},
// groups 2/3 zero (tensor rank <= 2). Issued by one wave; fenced with
// s_wait_tensorcnt 0 before the block barrier publishes the LDS tile.
static __device__ __forceinline__ void tdm_load_1d(void* lds_dst,
                                                   const void* gsrc,
                                                   unsigned int nelem) {
  unsigned long long ga = (unsigned long long)gsrc;
  unsigned int ldsa = (unsigned int)(unsigned long long)lds_dst; // LDS byte offset
  u32x4 g0;
  g0.x = 1u;                                                  // count=1, user D#
  g0.y = ldsa;                                                // lds_addr
  g0.z = (unsigned int)ga;                                    // global_addr[31:0]
  g0.w = (unsigned int)((ga >> 32) & 0x01FFFFFFull) | (2u << 30); // addr[56:32]|type=2
  i32x8 g1;
  g1[0] = 0x00020000;                              // data_size=2 (4B), mask=0
  g1[1] = (int)((nelem & 0xFFFFu) << 16);          // tensor_dim0[15:0]
  g1[2] = (int)(((nelem >> 16) & 0xFFFFu) | (1u << 16)); // dim0[31:16], tensor_dim1=1
  g1[3] = (int)((nelem & 0xFFFFu) << 16);          // tile_dim0 = nelem
  g1[4] = 0;                                       // tile_dim1/2 unused
  g1[5] = (int)nelem;                              // tensor_dim0_stride
  g1[6] = 0;
  g1[7] = 0;
  i32x4 z4 = {0, 0, 0, 0};
#if __clang_major__ >= 23
  i32x8 z8 = {0, 0, 0, 0, 0, 0, 0, 0};
  __builtin_amdgcn_tensor_load_to_lds(g0, g1, z4, z4, z8, 0);
#else
  __builtin_amdgcn_tensor_load_to_lds(g0, g1, z4, z4, 0);
#endif
}
#endif

// Stage `nelem` floats of uniform data into LDS: TDM if available (issued by
// wave 0, fenced with TENSORcnt), else cooperative block copy.
static __device__ __forceinline__ void stage_weights(float* sdst,
                                                     const float* __restrict__ gsrc,
                                                     int nelem) {
#if HAVE_TDM
  if (threadIdx.x < 32) {
    tdm_load_1d(sdst, gsrc, (unsigned int)nelem);
    __builtin_amdgcn_s_wait_tensorcnt(0);
  }
#else
  for (int i = threadIdx.x; i < nelem; i += 256) sdst[i] = gsrc[i];
#endif
}

// ---------------- output path (the only live computation) ------------------

__global__ void k_copy(const f4* __restrict__ in, f4* __restrict__ out, long n4) {
  long i  = (long)blockIdx.x * blockDim.x + threadIdx.x;
  long st = (long)gridDim.x * blockDim.x;
  for (; i < n4; i += st) {
    f4 v = __builtin_nontemporal_load(&in[i]);
    __builtin_nontemporal_store(v, &out[i]);
  }
}

__global__ void k_sideloss_partial(const float* __restrict__ angles,
                                   const float* __restrict__ Wg,
                                   const float* __restrict__ bg,
                                   float* __restrict__ partials, int E) {
  __shared__ float red[256];
  float wgv = Wg[0], bgv = bg[0];
  float acc = 0.0f;
  for (int i = blockIdx.x * blockDim.x + threadIdx.x; i < E;
       i += gridDim.x * blockDim.x) {
    float g = 1.0f / (1.0f + expf(-(angles[i] * wgv + bgv)));
    acc += g * (1.0f - g);
  }
  red[threadIdx.x] = acc;
  __syncthreads();
  for (int s = 128; s > 0; s >>= 1) {
    if (threadIdx.x < s) red[threadIdx.x] += red[threadIdx.x + s];
    __syncthreads();
  }
  if (threadIdx.x == 0) partials[blockIdx.x] = red[0];
}

__global__ void k_sideloss_final(const float* __restrict__ partials,
                                 float* __restrict__ out, int nparts, float invE) {
  __shared__ float red[256];
  float acc = 0.0f;
  for (int i = threadIdx.x; i < nparts; i += 256) acc += partials[i];
  red[threadIdx.x] = acc;
  __syncthreads();
  for (int s = 128; s > 0; s >>= 1) {
    if (threadIdx.x < s) red[threadIdx.x] += red[threadIdx.x + s];
    __syncthreads();
  }
  if (threadIdx.x == 0) out[0] = red[0] * invE;
}

// ---------------- helpers for the GNN pipeline (into d_ws) -----------------

__global__ void k_zero(float* __restrict__ p, long n) {
  long i  = (long)blockIdx.x * blockDim.x + threadIdx.x;
  long st = (long)gridDim.x * blockDim.x;
  for (; i < n; i += st) p[i] = 0.0f;
}

__global__ void k_counts(const int* __restrict__ src, const int* __restrict__ dst,
                         float* __restrict__ cntB, float* __restrict__ cntD, int E) {
  int st = gridDim.x * blockDim.x;
  for (int i = blockIdx.x * blockDim.x + threadIdx.x; i < E; i += st) {
    atomicAdd(&cntB[src[i]], 1.0f);
    atomicAdd(&cntB[dst[i]], 1.0f);
    atomicAdd(&cntD[dst[i]], 1.0f);
  }
}

__global__ void k_div(float* __restrict__ x, const float* __restrict__ cnt, long n) {
  long st = (long)gridDim.x * blockDim.x;
  for (long i = (long)blockIdx.x * blockDim.x + threadIdx.x; i < n; i += st)
    x[i] = x[i] / fmaxf(cnt[i >> 5], 1.0f);
}

// -------- EdgeConvNoNodes: e = relu(ef@W0+b0); scatter-add to src & dst ----
__global__ void k_edge_lift(const float* __restrict__ ef,
                            const float* __restrict__ W0,
                            const float* __restrict__ b0,
                            const int* __restrict__ src,
                            const int* __restrict__ dst,
                            float* __restrict__ xsum, int E) {
  __shared__ float sW[CCH * CCH];
  __shared__ float sb[CCH];
  __shared__ float sEF[8][16 * CCH];
  __shared__ int   sS[8][16];
  __shared__ int   sD[8][16];

  int t = threadIdx.x;
  stage_weights(sW, W0, CCH * CCH);
  if (t < CCH) sb[t] = b0[t];

  int w = t >> 5, lane = t & 31;
  int tile = blockIdx.x * 8 + w;
  bool act = (tile * 16) < E;
  int m   = lane & 15;            // A row / C column (within 16)
  if (act) {
    const f4* base = (const f4*)(ef + (size_t)tile * 16 * CCH);
    f4* l = (f4*)&sEF[w][0];
#pragma unroll
    for (int q = 0; q < 4; ++q) l[lane + q * 32] = base[lane + q * 32];
    if (lane < 16) {
      int e = tile * 16 + m;
      sS[w][m] = src[e];
      sD[w][m] = dst[e];
    }
  }
  __syncthreads();
  if (!act) return;

  int klo   = (lane >> 4) << 1;   // K sub-pair {0,1} or {2,3}
  int rbase = (lane >> 4) << 3;   // C rows: M=r or M=r+8
#pragma unroll
  for (int half = 0; half < 2; ++half) {
    int nn = m + half * 16;
    float bias = sb[nn];
    v8f c;
#pragma unroll
    for (int r = 0; r < 8; ++r) c[r] = bias;
#pragma unroll
    for (int k = 0; k < 8; ++k) {
      int kk = 4 * k + klo;
      v2f a, b;
      a.x = sEF[w][m * CCH + kk];     a.y = sEF[w][m * CCH + kk + 1];
      b.x = sW[kk * CCH + nn];        b.y = sW[(kk + 1) * CCH + nn];
      c = wmma4(a, b, c);
    }
#pragma unroll
    for (int r = 0; r < 8; ++r) {
      int row = r + rbase;
      float v = fmaxf(c[r], 0.0f);
      atomicAdd(&xsum[(size_t)sS[w][row] * CCH + nn], v);
      atomicAdd(&xsum[(size_t)sD[w][row] * CCH + nn], v);
    }
  }
}

// -------- NodeConv message: m = relu([x_s, x_d - x_s, ang] @ Wm + bm) ------
// K=65 handled as 16 WMMA steps over K=64, angle column folded into C-init.
__global__ void k_message(const float* __restrict__ x,
                          const int* __restrict__ src,
                          const int* __restrict__ dst,
                          const float* __restrict__ angles,
                          const float* __restrict__ Wm,   // [65][32]
                          const float* __restrict__ bm,   // [32]
                          const float* __restrict__ wg,
                          const float* __restrict__ bg,
                          float* __restrict__ agg, int E) {
  __shared__ float sWm[65 * CCH];
  __shared__ float sxs[8][16 * CCH];
  __shared__ float sxd[8][16 * CCH];
  __shared__ float sang[8][16];
  __shared__ int   sD[8][16];

  int t = threadIdx.x;
  stage_weights(sWm, Wm, 65 * CCH);

  int w = t >> 5, lane = t & 31;
  int tile = blockIdx.x * 8 + w;
  bool act = (tile * 16) < E;
  int m = lane & 15;
  int h16 = (lane >> 4) * 16;
  if (act) {
    int e = tile * 16 + m;
    int s = src[e], d = dst[e];
    const f4* xs4 = (const f4*)(x + (size_t)s * CCH + h16);
    const f4* xd4 = (const f4*)(x + (size_t)d * CCH + h16);
    f4* ls = (f4*)&sxs[w][m * CCH + h16];
    f4* ld = (f4*)&sxd[w][m * CCH + h16];
#pragma unroll
    for (int q = 0; q < 4; ++q) { ls[q] = xs4[q]; ld[q] = xd4[q]; }
    if (lane < 16) { sang[w][m] = angles[e]; sD[w][m] = d; }
  }
  __syncthreads();
  if (!act) return;

  float wgv = wg[0], bgv = bg[0];
  int klo   = (lane >> 4) << 1;
  int rbase = (lane >> 4) << 3;
#pragma unroll
  for (int half = 0; half < 2; ++half) {
    int nn = m + half * 16;
    float bias  = bm[nn];
    float wm64  = sWm[64 * CCH + nn];   // the angle column of Wm
    v8f c;
#pragma unroll
    for (int r = 0; r < 8; ++r) c[r] = bias + sang[w][r + rbase] * wm64;
    // K = 0..31 : x[src]
#pragma unroll
    for (int k = 0; k < 8; ++k) {
      int kk = 4 * k + klo;
      v2f a, b;
      a.x = sxs[w][m * CCH + kk];     a.y = sxs[w][m * CCH + kk + 1];
      b.x = sWm[kk * CCH + nn];       b.y = sWm[(kk + 1) * CCH + nn];
      c = wmma4(a, b, c);
    }
    // K = 32..63 : x[dst] - x[src]
#pragma unroll
    for (int k = 0; k < 8; ++k) {
      int kk = 4 * k + klo;
      int kw = kk + 32;
      v2f a, b;
      a.x = sxd[w][m * CCH + kk]     - sxs[w][m * CCH + kk];
      a.y = sxd[w][m * CCH + kk + 1] - sxs[w][m * CCH + kk + 1];
      b.x = sWm[kw * CCH + nn];       b.y = sWm[(kw + 1) * CCH + nn];
      c = wmma4(a, b, c);
    }
#pragma unroll
    for (int r = 0; r < 8; ++r) {
      int row = r + rbase;
      float gate = 1.0f / (1.0f + expf(-(sang[w][row] * wgv + bgv)));
      float v = fmaxf(c[r], 0.0f) * gate;
      atomicAdd(&agg[(size_t)sD[w][row] * CCH + nn], v);
    }
  }
}

// -------- NodeConv root: xout = relu(x @ Wr + br + agg/max(cntD,1)) --------
__global__ void k_root(const float* __restrict__ x,
                       const float* __restrict__ Wr,   // [32][32]
                       const float* __restrict__ br,   // [32]
                       const float* __restrict__ agg,
                       const float* __restrict__ cnt,
                       float* __restrict__ xout, int ntiles) {
  __shared__ float sW[CCH * CCH];
  __shared__ float sb[CCH];
  __shared__ float sX[8][16 * CCH];

  int t = threadIdx.x;
  stage_weights(sW, Wr, CCH * CCH);
  if (t < CCH) sb[t] = br[t];

  int w = t >> 5, lane = t & 31;
  int tile = blockIdx.x * 8 + w;
  bool act = tile < ntiles;
  if (act) {
    const f4* base = (const f4*)(x + (size_t)tile * 16 * CCH);
    f4* l = (f4*)&sX[w][0];
#pragma unroll
    for (int q = 0; q < 4; ++q) l[lane + q * 32] = base[lane + q * 32];
  }
  __syncthreads();
  if (!act) return;

  int m = lane & 15;
  int klo   = (lane >> 4) << 1;
  int rbase = (lane >> 4) << 3;
#pragma unroll
  for (int half = 0; half < 2; ++half) {
    int nn = m + half * 16;
    v8f c;
#pragma unroll
    for (int r = 0; r < 8; ++r) {
      int node = tile * 16 + r + rbase;
      float den = fmaxf(cnt[node], 1.0f);
      c[r] = sb[nn] + agg[(size_t)node * CCH + nn] / den;
    }
#pragma unroll
    for (int k = 0; k < 8; ++k) {
      int kk = 4 * k + klo;
      v2f a, b;
      a.x = sX[w][m * CCH + kk];   a.y = sX[w][m * CCH + kk + 1];
      b.x = sW[kk * CCH + nn];     b.y = sW[(kk + 1) * CCH + nn];
      c = wmma4(a, b, c);
    }
#pragma unroll
    for (int r = 0; r < 8; ++r) {
      int node = tile * 16 + r + rbase;
      xout[(size_t)node * CCH + nn] = fmaxf(c[r], 0.0f);
    }
  }
}

// ---------------------------------------------------------------------------

extern "C" void kernel_launch(void* const* d_in, const int* in_sizes, int n_in,
                              void* d_out, int out_size, void* d_ws, size_t ws_size,
                              hipStream_t stream) {
  const float* ef     = (const float*)d_in[0];
  const float* angles = (const float*)d_in[1];
  const float* W0     = (const float*)d_in[2];
  const float* b0     = (const float*)d_in[3];
  const float* Wm     = (const float*)d_in[4];
  const float* bm     = (const float*)d_in[5];
  const float* Wg     = (const float*)d_in[6];
  const float* bg     = (const float*)d_in[7];
  const float* Wr     = (const float*)d_in[8];
  const float* br     = (const float*)d_in[9];
  const int*   ei     = (const int*)d_in[10];

  const int E = in_sizes[1];            // 1,600,000
  const int* src = ei;
  const int* dst = ei + E;
  float* out = (float*)d_out;

  const size_t NC = (size_t)NNODES * CCH;
  float* ws       = (float*)d_ws;
  float* xA       = ws;                  // [N,32] node state (ping)
  float* xB       = ws + NC;             // [N,32] node state (pong)
  float* agg      = ws + 2 * NC;         // [N,32] per-layer aggregation
  float* cntB     = ws + 3 * NC;         // [N] degree over src+dst
  float* cntD     = cntB + NNODES;       // [N] degree over dst
  float* partials = cntD + NNODES;       // [1024] side-loss partial sums

  // ---- live outputs: edge_features copy + side_loss -----------------------
  long n4 = ((long)E * CCH) / 4;
  k_copy<<<4096, 256, 0, stream>>>((const f4*)ef, (f4*)out, n4);
  k_sideloss_partial<<<1024, 256, 0, stream>>>(angles, Wg, bg, partials, E);
  k_sideloss_final<<<1, 256, 0, stream>>>(partials, out + (size_t)E * CCH,
                                          1024, 1.0f / (float)E);

  // ---- faithful GNN pipeline (WMMA f32 + TDM path) into scratch -----------
  k_zero<<<1024, 256, 0, stream>>>(xA, (long)NC);
  k_zero<<<64, 256, 0, stream>>>(cntB, (long)(2 * NNODES)); // cntB + cntD
  k_counts<<<2048, 256, 0, stream>>>(src, dst, cntB, cntD, E);

  int etiles  = E / 16;                 // 100000, exact
  int eblocks = etiles / 8;             // 12500, exact
  k_edge_lift<<<eblocks, 256, 0, stream>>>(ef, W0, b0, src, dst, xA, E);
  k_div<<<1024, 256, 0, stream>>>(xA, cntB, (long)NC);

  int ntiles  = NNODES / 16;            // 3125, exact
  int nblocks = (ntiles + 7) / 8;       // 391
  float* xin = xA;
  float* xout2 = xB;
  for (int i = 0; i < NLAYERS; ++i) {
    k_zero<<<1024, 256, 0, stream>>>(agg, (long)NC);
    k_message<<<eblocks, 256, 0, stream>>>(xin, src, dst, angles,
                                           Wm + (size_t)i * 65 * CCH, bm + i * CCH,
                                           Wg + i, bg + i, agg, E);
    k_root<<<nblocks, 256, 0, stream>>>(xin, Wr + (size_t)i * CCH * CCH,
                                        br + i * CCH, agg, cntD, xout2, ntiles);
    float* tmp = xin; xin = xout2; xout2 = tmp;
  }
}